// get_loss_20023137534162
// MI455X (gfx1250) — compile-verified
//
#include <hip/hip_runtime.h>

typedef float v2f __attribute__((ext_vector_type(2)));
typedef float v8f __attribute__((ext_vector_type(8)));

#define BATCH  8
#define NPTS   4096
#define TILE   16
#define WAVES  8         // waves per block
#define TPB    (WAVES * 32)
#define CHUNK  2048      // targets staged in LDS per phase (32 KB)

// For each point in A (NA pts/batch), min squared distance to all points in Bp
// (NB pts/batch) via V_WMMA_F32_16X16X4_F32.
//
// WMMA computes D[n,m] = (px,py,pz,1) . (-2tx,-2ty,-2tz,t2) = t2[m] - 2 p.t,
// so min-dist[n] = p2[n] + min_m D[n,m].
//
// A block (8 waves) owns 8 consecutive 16-row tiles (all in one batch) and
// cooperatively stages target fragments {-2tx,-2ty,-2tz,t2} in LDS; the sweep
// loop is then ds_load_b64 + wmma + v_min per chunk.
__global__ __launch_bounds__(TPB)
void nn_min_kernel(const float* __restrict__ A, const float* __restrict__ Bp,
                   float* __restrict__ minout, int NA, int NB)
{
    __shared__ float4 stage[CHUNK];

    const int lane = threadIdx.x & 31;
    const int col  = lane & 15;          // row-in-tile (A frag) / column (B frag)
    const int hi   = lane >> 4;          // half select: K={0,1} vs K={2,3}
    const int wave = threadIdx.x >> 5;

    const int tile0 = blockIdx.x * WAVES;          // 8 tiles per block, same batch
    const int tilesPerBatch = NA / TILE;
    const int b    = tile0 / tilesPerBatch;
    const int tile = tile0 + wave;
    const int n0   = (tile - b * tilesPerBatch) * TILE;

    // --- A fragment (constant across the sweep) ---
    // lane l holds row n0+(l&15); lanes 0-15: K=0,1=(px,py); lanes 16-31: K=2,3=(pz,1)
    const float* pp = A + ((size_t)b * NA + (size_t)(n0 + col)) * 3;
    const float px = pp[0], py = pp[1], pz = pp[2];
    const float p2 = px * px + py * py + pz * pz;
    v2f afrag;
    afrag.x = hi ? pz   : px;
    afrag.y = hi ? 1.0f : py;

    v8f runmin;
#pragma unroll
    for (int i = 0; i < 8; ++i) runmin[i] = 3.0e38f;

    const float* bbase = Bp + (size_t)b * NB * 3;
    const float* ldsf  = (const float*)stage;
    const int laneOff  = col * 4 + hi * 2;   // float offset of this lane's b64 read

    for (int phase = 0; phase < NB; phase += CHUNK) {
        __syncthreads();   // all waves done reading previous phase
        // --- stage CHUNK targets: {-2tx,-2ty,-2tz,t2} per target ---
        for (int m = threadIdx.x; m < CHUNK; m += TPB) {
            const float* tp = bbase + (size_t)(phase + m) * 3;
            const float tx = tp[0], ty = tp[1], tz = tp[2];
            const float t2 = tx * tx + ty * ty + tz * tz;
            stage[m] = make_float4(-2.0f * tx, -2.0f * ty, -2.0f * tz, t2);
        }
        __syncthreads();

        // --- sweep staged targets in 16-column chunks ---
#pragma unroll 4
        for (int m0 = 0; m0 < CHUNK; m0 += TILE) {
            // lanes 0-15 read (-2tx,-2ty); lanes 16-31 read (-2tz,t2): conflict-free
            v2f bfrag = *(const v2f*)(ldsf + (size_t)m0 * 4 + laneOff);
            v8f czero = {};
            v8f d = __builtin_amdgcn_wmma_f32_16x16x4_f32(
                false, afrag, false, bfrag, (short)0, czero, false, false);
#pragma unroll
            for (int i = 0; i < 8; ++i) runmin[i] = fminf(runmin[i], d[i]);
        }
    }

    // --- finalize row mins ---
    // C/D layout: VGPR r, lanes 0-15 -> row M=r; lanes 16-31 -> row M=r+8.
    // Butterfly min over the 16 lanes of each half (xor masks < 16 stay in-half).
#pragma unroll
    for (int r = 0; r < 8; ++r) {
        float v = runmin[r];
        v = fminf(v, __shfl_xor(v, 1));
        v = fminf(v, __shfl_xor(v, 2));
        v = fminf(v, __shfl_xor(v, 4));
        v = fminf(v, __shfl_xor(v, 8));
        // storing lane chosen so that its locally-computed p2 (row n0+col)
        // matches the row this element represents:
        //   half0: row r     -> lane r       half1: row r+8 -> lane 24+r
        const int trow = r + (hi << 3);
        if (col == trow)
            minout[(size_t)b * NA + (size_t)(n0 + trow)] = v + p2;
    }
}

// Sum all per-point mins (both directions, NA==NB==NPTS), apply means, add
// lamda*bpp:  loss = sum/(B*N) + lamda[0]*bpp[0].
__global__ __launch_bounds__(256)
void reduce_loss_kernel(const float* __restrict__ mins, int total, float inv,
                        const float* __restrict__ bpp,
                        const float* __restrict__ lamda,
                        float* __restrict__ out)
{
    __shared__ float sdata[256];
    float s = 0.0f;
    for (int i = threadIdx.x; i < total; i += blockDim.x) s += mins[i];
    sdata[threadIdx.x] = s;
    __syncthreads();
#pragma unroll
    for (int off = 128; off > 0; off >>= 1) {
        if ((int)threadIdx.x < off) sdata[threadIdx.x] += sdata[threadIdx.x + off];
        __syncthreads();
    }
    if (threadIdx.x == 0)
        out[0] = sdata[0] * inv + lamda[0] * bpp[0];
}

extern "C" void kernel_launch(void* const* d_in, const int* in_sizes, int n_in,
                              void* d_out, int out_size, void* d_ws, size_t ws_size,
                              hipStream_t stream) {
    (void)in_sizes; (void)n_in; (void)out_size; (void)ws_size;
    const float* pred   = (const float*)d_in[0];   // [B, N, 3]
    const float* target = (const float*)d_in[1];   // [B, M, 3]
    const float* bpp    = (const float*)d_in[2];   // [1]
    const float* lamda  = (const float*)d_in[3];   // [1]
    float* out  = (float*)d_out;                   // scalar loss

    float* minx = (float*)d_ws;                    // [B*N] pred -> nearest target
    float* miny = minx + (size_t)BATCH * NPTS;     // [B*M] target -> nearest pred

    const int tiles = BATCH * (NPTS / TILE);       // 2048 wave-tiles per direction
    dim3 block(TPB);                               // 8 waves / block
    dim3 grid(tiles / WAVES);                      // 256 blocks

    nn_min_kernel<<<grid, block, 0, stream>>>(pred, target, minx, NPTS, NPTS);
    nn_min_kernel<<<grid, block, 0, stream>>>(target, pred, miny, NPTS, NPTS);

    const float inv = 1.0f / (float)((size_t)BATCH * NPTS);
    reduce_loss_kernel<<<1, 256, 0, stream>>>(minx, 2 * BATCH * NPTS, inv,
                                              bpp, lamda, out);
}